// R50D_LPro_CFFN_BN_SP_LA_P7777_CA_NK2468_R05_H2_N221_C64_E2_SACAV4_Lite_90864328114324
// MI455X (gfx1250) — compile-verified
//
#include <hip/hip_runtime.h>

// ---------------------------------------------------------------------------
// Patch cross-attention for MI455X (gfx1250, wave32, WMMA).
//
// Structure exploited: unfold params resolve to p=7, stride=21, dil=4, nk=12.
// Patch pixel positions 21*i + 4*a never collide, so fold is a pure scatter
// onto an 84x84 subgrid; all other pixels of the output are exactly bp[c].
//   Kernel 1: fill out with bias (134 MB stores -> the bandwidth floor).
//   Kernel 2: per (batch, patch, head): gather 49 px, all GEMMs via
//             v_wmma_f32_16x16x4_f32 in LDS, atomicAdd scatter of the
//             final Wp projection.
// ---------------------------------------------------------------------------

typedef float v2f __attribute__((ext_vector_type(2)));
typedef float v8f __attribute__((ext_vector_type(8)));

#define IC    64      // input channels
#define HW    65536   // 256*256
#define NKR   12      // patches per axis
#define SP    144     // patches per image
#define PP    49      // pixels per patch (7*7)
#define NP    64      // padded pixel count (WMMA tiles)
#define LP    65      // LDS row stride (floats), avoids bank conflicts
#define STRD  21
#define DIL   4
#define PKS   7

// LDS layout (floats). X buffers are dead after the conv stage, so KK and O
// alias them. Total = 14816 floats = 59264 bytes (< 64 KB static limit).
#define OFF_K1 0                       // 32*LP  : K1 head slice [32][64]
#define OFF_K2 (32 * LP)               // 32*LP  : K2 head slice
#define OFF_V1 (64 * LP)               // 16*LP  : V1 head slice [16][64]
#define OFF_V2 (64 * LP + 16 * LP)     // 16*LP  : V2 head slice
#define OFF_ST (64 * LP + 32 * LP)     // 256    : cmax|cinv|rmax|rinv
#define OFF_X1 (96 * LP + 256)         // 64*LP  : x1 patch [64ch][64px]
#define OFF_X2 (96 * LP + 256 + 64 * LP)
#define OFF_KK OFF_X1                  // 64*LP  : attention logits [64][64]
#define OFF_OO OFF_X2                  // 32*LP  : o1|o2 head outputs [32][64]
#define SMEM_F (96 * LP + 256 + 128 * LP)

__device__ __forceinline__ v8f wmma4(v2f a, v2f b, v8f c) {
  // D = A(16x4) * B(4x16) + C, fp32 in / fp32 accumulate.
  return __builtin_amdgcn_wmma_f32_16x16x4_f32(
      false, a, false, b, (short)0, c, false, false);
}

__global__ __launch_bounds__(256) void fill_bias_kernel(
    const float* __restrict__ bp, float* __restrict__ out) {
  const int plane = blockIdx.x;            // b*64 + c
  const float v = bp[plane & 63];
  const float4 vv = make_float4(v, v, v, v);
  float4* o4 = reinterpret_cast<float4*>(out + (size_t)plane * HW);
  for (int i = threadIdx.x; i < HW / 4; i += 256) o4[i] = vv;
}

__global__ __launch_bounds__(256) void patch_attn_kernel(
    const float* __restrict__ x1, const float* __restrict__ x2,
    const float* __restrict__ Wk, const float* __restrict__ bk,
    const float* __restrict__ Wv, const float* __restrict__ bv,
    const float* __restrict__ Wp, float* __restrict__ out) {
  __shared__ float sm[SMEM_F];

  const int tid  = threadIdx.x;
  const int lane = tid & 31;
  const int wid  = tid >> 5;
  const int half = lane >> 4;       // 0: lanes 0-15, 1: lanes 16-31
  const int l16  = lane & 15;

  int unit = blockIdx.x;            // [0, 8*144*2)
  const int h = unit & 1;           // head
  unit >>= 1;
  const int s = unit % SP;
  const int b = unit / SP;
  const int baseRow = (s / NKR) * STRD;
  const int baseCol = (s % NKR) * STRD;

  // ------------------- gather patch pixels (zero-padded 49->64) ------------
  for (int i = tid; i < IC * NP; i += 256) {
    const int c = i >> 6;
    const int pix = i & 63;
    float a1v = 0.f, a2v = 0.f;
    if (pix < PP) {
      const int gr = baseRow + (pix / PKS) * DIL;
      const int gc = baseCol + (pix % PKS) * DIL;
      const int g = ((b * IC + c) * 256 + gr) * 256 + gc;
      a1v = x1[g];
      a2v = x2[g];
    }
    sm[OFF_X1 + c * LP + pix] = a1v;
    sm[OFF_X2 + c * LP + pix] = a2v;
  }
  __syncthreads();

  // ------------------- stage 1: K/V projections (24 WMMA tile jobs) --------
  // K1h[32][64]=Wk[h*32+..]@X1+bk, K2h=Wk[64+h*32+..]@X2+bk,
  // V1h[16][64]=Wv[h*16+..]@X1+bv, V2h=Wv[32+h*16+..]@X2+bv.
  for (int t = wid; t < 24; t += 8) {
    int gemm, mi, ni;
    if (t < 16) { gemm = t >> 3; mi = (t >> 2) & 1; ni = t & 3; }
    else        { gemm = 2 + ((t - 16) >> 2); mi = 0; ni = (t - 16) & 3; }
    const float* Wg = (gemm < 2) ? Wk : Wv;
    const float* bg = (gemm < 2) ? bk : bv;
    int wrow0, xo, oo;
    switch (gemm) {
      case 0:  wrow0 = h * 32;      xo = OFF_X1; oo = OFF_K1; break;
      case 1:  wrow0 = 64 + h * 32; xo = OFF_X2; oo = OFF_K2; break;
      case 2:  wrow0 = h * 16;      xo = OFF_X1; oo = OFF_V1; break;
      default: wrow0 = 32 + h * 16; xo = OFF_X2; oo = OFF_V2; break;
    }
    v8f acc = {0.f, 0.f, 0.f, 0.f, 0.f, 0.f, 0.f, 0.f};
    const int arow = wrow0 + mi * 16 + l16;   // weight row
    const int bcol = ni * 16 + l16;           // pixel column
    for (int k0 = 0; k0 < 64; k0 += 4) {
      const int ka = k0 + half * 2;
      v2f a;  a.x  = Wg[arow * IC + ka];
              a.y  = Wg[arow * IC + ka + 1];
      v2f bb; bb.x = sm[xo + ka * LP + bcol];
              bb.y = sm[xo + (ka + 1) * LP + bcol];
      acc = wmma4(a, bb, acc);
    }
#pragma unroll
    for (int r = 0; r < 8; ++r) {
      const int row = mi * 16 + r + half * 8;
      sm[oo + row * LP + bcol] = acc[r] + bg[wrow0 + row];
    }
  }
  __syncthreads();

  // ------------------- stage 2: KK = scale * K1^T K2  (64x64, K=32) --------
  const float scale = 0.17677669529663687f;  // 32^-0.5
  for (int t = wid; t < 16; t += 8) {
    const int mi = t >> 2, ni = t & 3;
    v8f acc = {0.f, 0.f, 0.f, 0.f, 0.f, 0.f, 0.f, 0.f};
    const int arow = mi * 16 + l16;  // x (transposed read of K1)
    const int bcol = ni * 16 + l16;  // y
    for (int k0 = 0; k0 < 32; k0 += 4) {
      const int ka = k0 + half * 2;
      v2f a;  a.x  = sm[OFF_K1 + ka * LP + arow] * scale;
              a.y  = sm[OFF_K1 + (ka + 1) * LP + arow] * scale;
      v2f bb; bb.x = sm[OFF_K2 + ka * LP + bcol];
              bb.y = sm[OFF_K2 + (ka + 1) * LP + bcol];
      acc = wmma4(a, bb, acc);
    }
#pragma unroll
    for (int r = 0; r < 8; ++r)
      sm[OFF_KK + (mi * 16 + r + half * 8) * LP + bcol] = acc[r];
  }
  __syncthreads();

  // ------------------- stage 3: softmax statistics -------------------------
  if (tid < 64) {                 // column-wise (softmax over x, for a1)
    const int y = tid;
    float mx = -1e30f;
    for (int x = 0; x < PP; ++x) mx = fmaxf(mx, sm[OFF_KK + x * LP + y]);
    float ss = 0.f;
    for (int x = 0; x < PP; ++x) ss += __expf(sm[OFF_KK + x * LP + y] - mx);
    sm[OFF_ST + y]      = mx;
    sm[OFF_ST + 64 + y] = 1.0f / ss;
  } else if (tid < 128) {         // row-wise (softmax over y, for a2)
    const int x = tid - 64;
    float mx = -1e30f;
    for (int y = 0; y < PP; ++y) mx = fmaxf(mx, sm[OFF_KK + x * LP + y]);
    float ss = 0.f;
    for (int y = 0; y < PP; ++y) ss += __expf(sm[OFF_KK + x * LP + y] - mx);
    sm[OFF_ST + 128 + x] = mx;
    sm[OFF_ST + 192 + x] = 1.0f / ss;
  }
  __syncthreads();

  // ------------- stage 4: O1 = V1 @ A1 ; O2 = V2 @ A2^T (one tile/wave) ----
  // Softmax weights are materialized on the fly inside the B fragment.
  {
    const int isO2 = wid >> 2;
    const int ni = wid & 3;
    const int vo = isO2 ? OFF_V2 : OFF_V1;
    const int bcol = ni * 16 + l16;
    v8f acc = {0.f, 0.f, 0.f, 0.f, 0.f, 0.f, 0.f, 0.f};
    for (int k0 = 0; k0 < 64; k0 += 4) {
      const int ka = k0 + half * 2;
      v2f a;  a.x = sm[vo + l16 * LP + ka];
              a.y = sm[vo + l16 * LP + ka + 1];
      v2f bb;
      if (!isO2) {  // A1[x][y] = exp(kk[x][y]-cmax[y])*cinv[y], 0 for x>=49
        const float m0 = sm[OFF_ST + bcol], iv = sm[OFF_ST + 64 + bcol];
        bb.x = (ka < PP)     ? __expf(sm[OFF_KK + ka * LP + bcol] - m0) * iv : 0.f;
        bb.y = (ka + 1 < PP) ? __expf(sm[OFF_KK + (ka + 1) * LP + bcol] - m0) * iv : 0.f;
      } else {      // A2T[y][x] = exp(kk[x][y]-rmax[x])*rinv[x], 0 for y>=49
        const float m0 = sm[OFF_ST + 128 + bcol], iv = sm[OFF_ST + 192 + bcol];
        bb.x = (ka < PP)     ? __expf(sm[OFF_KK + bcol * LP + ka] - m0) * iv : 0.f;
        bb.y = (ka + 1 < PP) ? __expf(sm[OFF_KK + bcol * LP + ka + 1] - m0) * iv : 0.f;
      }
      acc = wmma4(a, bb, acc);
    }
#pragma unroll
    for (int r = 0; r < 8; ++r)
      sm[OFF_OO + (r + half * 8 + isO2 * 16) * LP + bcol] = acc[r];
  }
  __syncthreads();

  // ------------- stage 5: partial projection P = Wp[:,headcols] @ O --------
  // Local O rows r<16 -> concat channel h*16+r (o1); r>=16 -> 32+h*16+(r-16).
  for (int t = wid; t < 16; t += 8) {
    const int mi = t >> 2, ni = t & 3;
    v8f acc = {0.f, 0.f, 0.f, 0.f, 0.f, 0.f, 0.f, 0.f};
    const int arow = mi * 16 + l16;  // output channel
    const int bcol = ni * 16 + l16;  // pixel
    for (int k0 = 0; k0 < 32; k0 += 4) {
      const int ka = k0 + half * 2;
      const int c0 = (ka < 16)     ? (h * 16 + ka)          : (16 + h * 16 + ka);
      const int c1 = (ka + 1 < 16) ? (h * 16 + ka + 1)      : (17 + h * 16 + ka);
      v2f a;  a.x  = Wp[arow * 64 + c0];
              a.y  = Wp[arow * 64 + c1];
      v2f bb; bb.x = sm[OFF_OO + ka * LP + bcol];
              bb.y = sm[OFF_OO + (ka + 1) * LP + bcol];
      acc = wmma4(a, bb, acc);
    }
    if (bcol < PP) {
      const int gr = baseRow + (bcol / PKS) * DIL;
      const int gc = baseCol + (bcol % PKS) * DIL;
#pragma unroll
      for (int r = 0; r < 8; ++r) {
        const int oc = mi * 16 + r + half * 8;
        atomicAdd(&out[((b * 64 + oc) * 256 + gr) * 256 + gc], acc[r]);
      }
    }
  }
}

extern "C" void kernel_launch(void* const* d_in, const int* in_sizes, int n_in,
                              void* d_out, int out_size, void* d_ws, size_t ws_size,
                              hipStream_t stream) {
  (void)in_sizes; (void)n_in; (void)d_ws; (void)ws_size; (void)out_size;
  const float* x1 = (const float*)d_in[0];
  const float* x2 = (const float*)d_in[1];
  const float* Wk = (const float*)d_in[2];
  const float* bk = (const float*)d_in[3];
  const float* Wv = (const float*)d_in[4];
  const float* bv = (const float*)d_in[5];
  const float* Wp = (const float*)d_in[6];
  const float* bp = (const float*)d_in[7];
  float* out = (float*)d_out;

  // 1) out[b,c,:,:] = bp[c] everywhere (the fold covers only 84x84 pixels).
  fill_bias_kernel<<<8 * 64, 256, 0, stream>>>(bp, out);
  // 2) per (batch, patch, head) fused attention + projection, atomic scatter.
  patch_attn_kernel<<<8 * SP * 2, 256, 0, stream>>>(x1, x2, Wk, bk, Wv, bv, Wp, out);
}